// SO3Layer_79182017069775
// MI455X (gfx1250) — compile-verified
//
#include <hip/hip_runtime.h>
#include <math.h>

typedef __attribute__((ext_vector_type(2))) float v2f;
typedef __attribute__((ext_vector_type(8))) float v8f;

#define WIDTH 256
#define WPITCH 260          // LDS pitch for padded weights: banks = nl*4+k (conflict-free)
#define YPITCH 12           // LDS pitch for 3x3 staging (conflict-free for 16 lanes)
#define WAVES_PER_BLOCK 4
#define ROWS_PER_WAVE 16
#define ROWS_PER_BLOCK (WAVES_PER_BLOCK * ROWS_PER_WAVE)   // 64

// ---- 3x3 symmetric Jacobi rotation on pair (P,Q), third index R implied ----
template <int P, int Q>
static __device__ __forceinline__ void jrot(float& app, float& aqq, float& apq,
                                            float& arp, float& arq, float v[9]) {
    float c = 1.0f, s = 0.0f, t = 0.0f;
    if (fabsf(apq) > 1e-20f) {
        float tau = (aqq - app) * 0.5f / apq;
        t = copysignf(1.0f, tau) / (fabsf(tau) + sqrtf(1.0f + tau * tau));
        c = rsqrtf(1.0f + t * t);
        s = t * c;
    }
    float d = t * apq;
    app -= d;
    aqq += d;
    apq = 0.0f;
    float rp = arp, rq = arq;
    arp = c * rp - s * rq;
    arq = s * rp + c * rq;
#pragma unroll
    for (int i = 0; i < 3; ++i) {
        float vp = v[i * 3 + P], vq = v[i * 3 + Q];
        v[i * 3 + P] = c * vp - s * vq;
        v[i * 3 + Q] = s * vp + c * vq;
    }
}

// Nearest rotation (special-orthogonal Procrustes) of row-major 3x3 m -> r.
// Equivalent to U diag(1,1,det(UV^T)) V^T from the SVD of m.
static __device__ __forceinline__ void nearest_rotation(const float m[9], float r[9]) {
    // A = m^T m (symmetric PSD)
    float a00 = m[0] * m[0] + m[3] * m[3] + m[6] * m[6];
    float a01 = m[0] * m[1] + m[3] * m[4] + m[6] * m[7];
    float a02 = m[0] * m[2] + m[3] * m[5] + m[6] * m[8];
    float a11 = m[1] * m[1] + m[4] * m[4] + m[7] * m[7];
    float a12 = m[1] * m[2] + m[4] * m[5] + m[7] * m[8];
    float a22 = m[2] * m[2] + m[5] * m[5] + m[8] * m[8];
    float v[9] = {1.f, 0.f, 0.f, 0.f, 1.f, 0.f, 0.f, 0.f, 1.f};
#pragma unroll
    for (int sw = 0; sw < 5; ++sw) {
        jrot<0, 1>(a00, a11, a01, a02, a12, v);
        jrot<0, 2>(a00, a22, a02, a01, a12, v);
        jrot<1, 2>(a11, a22, a12, a01, a02, v);
    }
    // Sort eigenpairs descending (column 2 = smallest singular direction)
    float w0 = a00, w1 = a11, w2 = a22;
    if (w0 < w1) {
        float t = w0; w0 = w1; w1 = t;
#pragma unroll
        for (int i = 0; i < 3; ++i) { float x = v[i*3+0]; v[i*3+0] = v[i*3+1]; v[i*3+1] = x; }
    }
    if (w1 < w2) {
        float t = w1; w1 = w2; w2 = t;
#pragma unroll
        for (int i = 0; i < 3; ++i) { float x = v[i*3+1]; v[i*3+1] = v[i*3+2]; v[i*3+2] = x; }
    }
    if (w0 < w1) {
        float t = w0; w0 = w1; w1 = t;
#pragma unroll
        for (int i = 0; i < 3; ++i) { float x = v[i*3+0]; v[i*3+0] = v[i*3+1]; v[i*3+1] = x; }
    }
    float v0x = v[0], v0y = v[3], v0z = v[6];
    float v1x = v[1], v1y = v[4], v1z = v[7];
    // u0 = normalize(M v0)
    float u0x = m[0] * v0x + m[1] * v0y + m[2] * v0z;
    float u0y = m[3] * v0x + m[4] * v0y + m[5] * v0z;
    float u0z = m[6] * v0x + m[7] * v0y + m[8] * v0z;
    float inv = rsqrtf(fmaxf(u0x * u0x + u0y * u0y + u0z * u0z, 1e-30f));
    u0x *= inv; u0y *= inv; u0z *= inv;
    // u1 = normalize(M v1 - proj_u0)
    float t1x = m[0] * v1x + m[1] * v1y + m[2] * v1z;
    float t1y = m[3] * v1x + m[4] * v1y + m[5] * v1z;
    float t1z = m[6] * v1x + m[7] * v1y + m[8] * v1z;
    float dp = u0x * t1x + u0y * t1y + u0z * t1z;
    t1x -= dp * u0x; t1y -= dp * u0y; t1z -= dp * u0z;
    inv = rsqrtf(fmaxf(t1x * t1x + t1y * t1y + t1z * t1z, 1e-30f));
    float u1x = t1x * inv, u1y = t1y * inv, u1z = t1z * inv;
    // u2 = u0 x u1 ; v2 = v0 x v1  (right-handed frames -> det flip handled implicitly)
    float u2x = u0y * u1z - u0z * u1y;
    float u2y = u0z * u1x - u0x * u1z;
    float u2z = u0x * u1y - u0y * u1x;
    float v2x = v0y * v1z - v0z * v1y;
    float v2y = v0z * v1x - v0x * v1z;
    float v2z = v0x * v1y - v0y * v1x;
    r[0] = u0x * v0x + u1x * v1x + u2x * v2x;
    r[1] = u0x * v0y + u1x * v1y + u2x * v2y;
    r[2] = u0x * v0z + u1x * v1z + u2x * v2z;
    r[3] = u0y * v0x + u1y * v1x + u2y * v2x;
    r[4] = u0y * v0y + u1y * v1y + u2y * v2y;
    r[5] = u0y * v0z + u1y * v1z + u2y * v2z;
    r[6] = u0z * v0x + u1z * v1x + u2z * v2x;
    r[7] = u0z * v0y + u1z * v1y + u2z * v2y;
    r[8] = u0z * v0z + u1z * v1z + u2z * v2z;
}

__global__ __launch_bounds__(128) void SO3Layer_79182017069775_kernel(
    const float* __restrict__ x, const float* __restrict__ fcw,
    const float* __restrict__ fcb, float* __restrict__ out) {
    __shared__ float wlds[16 * WPITCH];                              // fc_w padded to 16 rows
    __shared__ float blds[16];                                       // bias padded
    __shared__ float ylds[WAVES_PER_BLOCK][ROWS_PER_WAVE][YPITCH];   // 3x3 staging

    const int tid = threadIdx.x;
    // Stage padded weight matrix (rows 9..15 zero) and bias into LDS.
    for (int i = tid; i < 16 * WIDTH; i += 128) {
        int n = i >> 8;
        int k = i & (WIDTH - 1);
        wlds[n * WPITCH + k] = (n < 9) ? fcw[n * WIDTH + k] : 0.0f;
    }
    if (tid < 16) blds[tid] = (tid < 9) ? fcb[tid] : 0.0f;
    __syncthreads();

    const int wave = tid >> 5;
    const int lane = tid & 31;
    const int nl   = lane & 15;               // M for A-frag / N for B-frag
    const int koff = (lane >> 4) << 1;        // K sub-offset per ISA 16x4 f32 layout
    const int rowBase = blockIdx.x * ROWS_PER_BLOCK + wave * ROWS_PER_WAVE;

    const float* xrow = x + (size_t)(rowBase + nl) * WIDTH + koff;
    const float* wrow = wlds + nl * WPITCH + koff;

    v8f acc = {0.f, 0.f, 0.f, 0.f, 0.f, 0.f, 0.f, 0.f};
#pragma unroll 8
    for (int k = 0; k < WIDTH; k += 4) {
        v2f a = *(const v2f*)(xrow + k);      // A frag: lane holds M=nl, K=k+koff..+1
        v2f b = *(const v2f*)(wrow + k);      // B frag: lane holds N=nl, K=k+koff..+1
        acc = __builtin_amdgcn_wmma_f32_16x16x4_f32(
            /*neg_a=*/false, a, /*neg_b=*/false, b,
            /*c_mod=*/(short)0, acc, /*reuse_a=*/false, /*reuse_b=*/false);
    }

    // Scatter D tile (+bias) to LDS so each lane can gather one full 3x3 row-matrix.
    const int mbase = (lane >> 4) << 3;       // VGPR i holds M = mbase + i
    if (nl < 9) {
        float bias = blds[nl];
#pragma unroll
        for (int i = 0; i < 8; ++i)
            ylds[wave][mbase + i][nl] = acc[i] + bias;
    }
    __syncthreads();

    if (lane < 16) {
        float mm[9], rr[9];
#pragma unroll
        for (int j = 0; j < 9; ++j) mm[j] = ylds[wave][lane][j];
        nearest_rotation(mm, rr);
        float* op = out + (size_t)(rowBase + lane) * 9;
#pragma unroll
        for (int j = 0; j < 9; ++j) op[j] = rr[j];
    }
}

extern "C" void kernel_launch(void* const* d_in, const int* in_sizes, int n_in,
                              void* d_out, int out_size, void* d_ws, size_t ws_size,
                              hipStream_t stream) {
    const float* x   = (const float*)d_in[0];
    const float* fcw = (const float*)d_in[1];
    const float* fcb = (const float*)d_in[2];
    float* out = (float*)d_out;
    const int n = in_sizes[0] / WIDTH;                 // 262144 points
    const int grid = n / ROWS_PER_BLOCK;               // 4096 blocks of 64 rows
    SO3Layer_79182017069775_kernel<<<dim3(grid), dim3(128), 0, stream>>>(x, fcw, fcb, out);
}